// MVPNet3D_78159814852892
// MI455X (gfx1250) — compile-verified
//
#include <hip/hip_runtime.h>

typedef __attribute__((ext_vector_type(2))) float v2f;
typedef __attribute__((ext_vector_type(8))) float v8f;

#define NPTS   32768
#define NBATCH 4
#define P2     48      // LDS pitch in float2 units: 48*8B = 32 banks (mod 64) between
                       // adjacent kpair rows -> the two half-waves of a B load are
                       // bank-disjoint; also exactly the 48 gemm columns (no padding).

// One 16(M) x 48(cols) output strip: 3 column tiles, K = ksteps*4.
// LDS activation layout: X[kpair][col] as float2 = (k=2*kpair, k=2*kpair+1).
// A-operand (16x4 f32): lane<16: rows m0+lane, K=kb..kb+1 ; lane>=16: K=kb+2..kb+3
// B-operand (4x16 f32): one ds_load_b64 of X[2*ks+half][ct*16+lc].
__device__ __forceinline__ void gemm_tile(
    const float* __restrict__ W, int ldw,
    const v2f* __restrict__ Xl, v2f* __restrict__ Yl,
    const float* __restrict__ sc, const float* __restrict__ sh,
    int m0, int ksteps, int lane)
{
    const int half = lane >> 4;
    const int lc   = lane & 15;

    v8f acc[3] = {};
    const float* wbase = W + (m0 + lc) * ldw + 2 * half;   // 8B-aligned for ldw=68/128/64

    for (int ks = 0; ks < ksteps; ++ks) {
        const v2f a = *reinterpret_cast<const v2f*>(wbase + 4 * ks);   // global_load_b64 (L2-hit)
        const v2f* xb = Xl + (2 * ks + half) * P2 + lc;
#pragma unroll
        for (int ct = 0; ct < 3; ++ct) {
            const v2f bv = xb[ct * 16];                                // ds_load_b64
            acc[ct] = __builtin_amdgcn_wmma_f32_16x16x4_f32(
                false, a, false, bv, (short)0, acc[ct], false, false);
        }
    }

    // Fused bias+BN affine + ReLU; write channel pairs -> ds_store_b64, paired layout
    // D layout: VGPR r -> row m0 + r + 8*half, col ct*16 + lc
    const int rbase = (m0 >> 1) + 4 * half;
    const int mbase = m0 + 8 * half;
#pragma unroll
    for (int ct = 0; ct < 3; ++ct) {
#pragma unroll
        for (int i = 0; i < 4; ++i) {
            v2f y;
            y.x = fmaxf(fmaf(acc[ct][2 * i],     sc[mbase + 2 * i],     sh[mbase + 2 * i]),     0.0f);
            y.y = fmaxf(fmaf(acc[ct][2 * i + 1], sc[mbase + 2 * i + 1], sh[mbase + 2 * i + 1]), 0.0f);
            Yl[(rbase + i) * P2 + ct * 16 + lc] = y;
        }
    }
}

__global__ __launch_bounds__(128)
void mvpnet_fused(const float* __restrict__ src, const float* __restrict__ tgt,
                  const float* __restrict__ feat,
                  const float* __restrict__ W1, const float* __restrict__ b1,
                  const float* __restrict__ g1, const float* __restrict__ be1,
                  const float* __restrict__ W2, const float* __restrict__ b2,
                  const float* __restrict__ g2, const float* __restrict__ be2,
                  const float* __restrict__ W3, const float* __restrict__ b3,
                  const float* __restrict__ g3, const float* __restrict__ be3,
                  float* __restrict__ out)
{
    __shared__ v2f smem[6496];             // 51.97 KB
    v2f* X0 = smem;                        // 34 kpairs * 48 = 1632 (68 rows; reused as Y3)
    v2f* Y1 = X0 + 1632;                   // 64 kpairs * 48 = 3072 (128 rows)
    v2f* Y2 = Y1 + 3072;                   // 32 kpairs * 48 = 1536 (64 rows)
    v2f* Y3 = X0;                          // 32 kpairs needed <= 34 available
    float* sc1 = reinterpret_cast<float*>(Y2 + 1536);   // 128
    float* sh1 = sc1 + 128;                // 128
    float* sc2 = sh1 + 128;                // 64
    float* sh2 = sc2 + 64;                 // 64
    float* sc3 = sh2 + 64;                 // 64
    float* sh3 = sc3 + 64;                 // 64

    const int tid  = threadIdx.x;
    const int wave = tid >> 5;
    const int lane = tid & 31;
    const int blk  = blockIdx.x;
    const int b    = blk >> 11;            // / (32768/16)
    const int n0   = (blk & 2047) << 4;    // * 16

    const float inv_s = rsqrtf(1.0f + 1e-5f);

    // Fold bias + eval-BN into per-channel affine: y = scale*acc + shift
    for (int i = tid; i < 256; i += 128) {
        if (i < 128)      { sc1[i] = g1[i] * inv_s; sh1[i] = fmaf(b1[i] * inv_s, g1[i], be1[i]); }
        else if (i < 192) { int m = i - 128; sc2[m] = g2[m] * inv_s; sh2[m] = fmaf(b2[m] * inv_s, g2[m], be2[m]); }
        else              { int m = i - 192; sc3[m] = g3[m] * inv_s; sh3[m] = fmaf(b3[m] * inv_s, g3[m], be3[m]); }
    }

    // Stage feature[:, :, n0:n0+16, 0:3] -> rows 0..63 of X0 (paired layout, col = 3*p + k)
    for (int i = tid; i < 1024; i += 128) {
        const int p = i & 15, c = i >> 4;
        const size_t off = ((size_t)(b * 64 + c) * NPTS + (size_t)(n0 + p)) * 16;
        const float4 f4 = *reinterpret_cast<const float4*>(feat + off);   // 64B-aligned row
        float* xp = reinterpret_cast<float*>(X0 + (c >> 1) * P2 + 3 * p) + (c & 1);
        xp[0] = f4.x; xp[2] = f4.y; xp[4] = f4.z;
    }

    // Relation channels: rows 64..66 = diff_xyz, row 67 = squared distance
    // Paired: kpair 32 = (diff_x, diff_y), kpair 33 = (diff_z, dist)
    if (tid < 16) {
        const int p = tid, n = n0 + p;
        float tg[3], sv[3][3];
#pragma unroll
        for (int ax = 0; ax < 3; ++ax) {
            tg[ax] = tgt[(size_t)(b * 3 + ax) * NPTS + n];
            const size_t off = ((size_t)(b * 3 + ax) * NPTS + (size_t)n) * 16;
            const float4 s4 = *reinterpret_cast<const float4*>(src + off);
            sv[ax][0] = s4.x; sv[ax][1] = s4.y; sv[ax][2] = s4.z;
        }
#pragma unroll
        for (int k = 0; k < 3; ++k) {
            const float e0 = sv[0][k] - tg[0];
            const float e1 = sv[1][k] - tg[1];
            const float e2 = sv[2][k] - tg[2];
            const float d  = fmaf(e0, e0, fmaf(e1, e1, e2 * e2));
            v2f lo; lo.x = e0; lo.y = e1;
            v2f hi; hi.x = e2; hi.y = d;
            X0[32 * P2 + 3 * p + k] = lo;
            X0[33 * P2 + 3 * p + k] = hi;
        }
    }
    __syncthreads();

    // Layer 1: (128x68) @ (68x48) ; each wave owns two 16-row M tiles, K = 17 steps of 4
    gemm_tile(W1, 68, X0, Y1, sc1, sh1, 32 * wave,      17, lane);
    gemm_tile(W1, 68, X0, Y1, sc1, sh1, 32 * wave + 16, 17, lane);
    __syncthreads();

    // Layer 2: (64x128) @ (128x48) ; one M tile per wave, K = 32 steps
    gemm_tile(W2, 128, Y1, Y2, sc2, sh2, 16 * wave, 32, lane);
    __syncthreads();

    // Layer 3: (64x64) @ (64x48) ; one M tile per wave, K = 16 steps (Y3 aliases X0, dead since L1)
    gemm_tile(W3, 64, Y2, Y3, sc3, sh3, 16 * wave, 16, lane);
    __syncthreads();

    // Neighbor-sum reduction + coalesced store: out[b, c, n0+p]
    for (int i = tid; i < 1024; i += 128) {
        const int p = i & 15, c = i >> 4;
        const float* yp = reinterpret_cast<const float*>(Y3 + (c >> 1) * P2 + 3 * p) + (c & 1);
        out[(size_t)(b * 64 + c) * NPTS + (size_t)(n0 + p)] = yp[0] + yp[2] + yp[4];
    }
}

extern "C" void kernel_launch(void* const* d_in, const int* in_sizes, int n_in,
                              void* d_out, int out_size, void* d_ws, size_t ws_size,
                              hipStream_t stream) {
    (void)in_sizes; (void)n_in; (void)d_ws; (void)ws_size; (void)out_size;
    const float* src  = (const float*)d_in[0];
    const float* tgt  = (const float*)d_in[1];
    const float* feat = (const float*)d_in[2];
    const float* W1   = (const float*)d_in[3];
    const float* b1   = (const float*)d_in[4];
    const float* g1   = (const float*)d_in[5];
    const float* be1  = (const float*)d_in[6];
    const float* W2   = (const float*)d_in[7];
    const float* b2   = (const float*)d_in[8];
    const float* g2   = (const float*)d_in[9];
    const float* be2  = (const float*)d_in[10];
    const float* W3   = (const float*)d_in[11];
    const float* b3   = (const float*)d_in[12];
    const float* g3   = (const float*)d_in[13];
    const float* be3  = (const float*)d_in[14];
    float* out = (float*)d_out;

    const int grid = NBATCH * (NPTS / 16);   // 8192 workgroups, 4 waves each
    mvpnet_fused<<<grid, 128, 0, stream>>>(src, tgt, feat,
                                           W1, b1, g1, be1,
                                           W2, b2, g2, be2,
                                           W3, b3, g3, be3,
                                           out);
}